// HD_Segformer_88373247083160
// MI455X (gfx1250) — compile-verified
//
#include <hip/hip_runtime.h>
#include <math.h>

typedef __attribute__((ext_vector_type(2))) float v2f;
typedef __attribute__((ext_vector_type(8))) float v8f;

#define NCLS 11

__device__ __forceinline__ unsigned f2ord(float f) {
    unsigned u = __float_as_uint(f);
    return (u & 0x80000000u) ? ~u : (u | 0x80000000u);
}

__global__ void init_keys_kernel(unsigned long long* keys) {
    if (threadIdx.x < 4) keys[threadIdx.x] = 0ULL;
}

// ---------------- Stage 1: 4x4 avgpool + first_masks (WMMA f32 16x16x4) + argmax ----
__global__ __launch_bounds__(256) void stage1_kernel(
    const float* __restrict__ im, const float* __restrict__ W1,
    const float* __restrict__ b1, float* __restrict__ first_masks,
    unsigned long long* __restrict__ keys)
{
    __shared__ float sr[256], sg[256], sb[256];
    const int t   = threadIdx.x;
    const int gid = blockIdx.x * 256 + t;          // 0 .. 65535  (4*128*128)
    const int b   = gid >> 14;
    const int pix = gid & 16383;
    const int py  = pix >> 7, px = pix & 127;

    // 4x4 mean pool of im
    const float* base = im + (size_t)b * 3u * 512u * 512u;
    float pr = 0.f, pg = 0.f, pb = 0.f;
    {
        const int y0 = py * 4, x0 = px * 4;
        float acc[3];
        #pragma unroll
        for (int c = 0; c < 3; ++c) {
            const float* p = base + (size_t)c * 512u * 512u + (size_t)y0 * 512u + x0;
            float s = 0.f;
            #pragma unroll
            for (int i = 0; i < 4; ++i)
                #pragma unroll
                for (int j = 0; j < 4; ++j)
                    s += p[i * 512 + j];
            acc[c] = s * (1.0f / 16.0f);
        }
        pr = acc[0]; pg = acc[1]; pb = acc[2];
    }
    sr[t] = pr; sg[t] = pg; sb[t] = pb;

    // score = sum_{k=1..10} (W1[k]·pooled + b1[k])  -> collapses to one dot product
    {
        float s0 = 0.f, s1 = 0.f, s2 = 0.f, sbias = 0.f;
        #pragma unroll
        for (int k = 1; k < NCLS; ++k) {
            s0 += W1[k * 3 + 0]; s1 += W1[k * 3 + 1]; s2 += W1[k * 3 + 2];
            sbias += b1[k];
        }
        const float score = s0 * pr + s1 * pg + s2 * pb + sbias;
        const unsigned long long key =
            ((unsigned long long)f2ord(score) << 32) |
            (unsigned long long)(0xFFFFFFFFu - (unsigned)pix);
        atomicMax(&keys[b], key);
    }
    __syncthreads();

    // WMMA F32 16x16x4:  D(16 pix x 16 cls) = A(16 pix x [r,g,b,1]) * B([W1^T; b1] 4x16)
    const int lane = t & 31;
    const int w    = t >> 5;
    const int m    = lane & 15;   // A: pixel row / B,D: column (class)
    const int h    = lane >> 4;   // lane half
    v2f bop;
    if (m < NCLS) {
        bop.x = h ? W1[m * 3 + 2] : W1[m * 3 + 0];   // K=2 : K=0
        bop.y = h ? b1[m]         : W1[m * 3 + 1];   // K=3 : K=1
    } else {
        bop.x = 0.f; bop.y = 0.f;
    }
    #pragma unroll
    for (int grp = 0; grp < 2; ++grp) {
        const int lb = w * 32 + grp * 16;            // LDS base of this 16-pixel group
        v2f a;
        a.x = h ? sb[lb + m] : sr[lb + m];           // K=2 : K=0
        a.y = h ? 1.0f       : sg[lb + m];           // K=3 : K=1  (1.0 folds in bias)
        v8f c = {};
        v8f d = __builtin_amdgcn_wmma_f32_16x16x4_f32(
            false, a, false, bop, (short)0, c, false, false);
        if (m < NCLS) {
            #pragma unroll
            for (int j = 0; j < 8; ++j) {
                const int pl = lb + j + 8 * h;               // pixel row M
                const int gp = blockIdx.x * 256 + pl;
                const int bb = gp >> 14, pp = gp & 16383;
                first_masks[((size_t)bb * NCLS + m) * 16384u + pp] = d[j];
            }
        }
    }
}

// ---------------- fused retina helpers --------------------------------------
__device__ __forceinline__ void ring_gather(
    const float* __restrict__ il, int k, float C,
    float fy16, float fx16, float dyf2, float dxf2, float rad2,
    float& r, float& g, float& bl)
{
    const float s  = ((float)k * rad2 - C) / rad2;
    const int gy = (int)floorf(fy16 + dyf2 * s);
    const int gx = (int)floorf(fx16 + dxf2 * s);
    const int H  = 2049 - k;                        // pooled size: 2048 - k + 1
    if (gy >= 0 && gy < H && gx >= 0 && gx < H) {
        const float inv = 1.0f / (float)(k * k);
        float acc[3];
        #pragma unroll
        for (int c = 0; c < 3; ++c) {
            const float* p = il + (size_t)c * 2048u * 2048u + (size_t)gy * 2048u + gx;
            float sum = 0.f;
            for (int i = 0; i < k; ++i)
                for (int j = 0; j < k; ++j)
                    sum += p[(size_t)i * 2048u + j];
            acc[c] = sum * inv;
        }
        r = acc[0]; g = acc[1]; bl = acc[2];
    } else {
        r = 0.f; g = 0.f; bl = 0.f;
    }
}

// ---------------- Stage 3: bilinear masks + fused retina/hd + softmax -------
__global__ __launch_bounds__(256) void stage3_kernel(
    const float* __restrict__ im_large,
    const float* __restrict__ first_masks,
    const unsigned long long* __restrict__ keys,
    const float* __restrict__ W2, const float* __restrict__ b2,
    float* __restrict__ out)
{
    const int gid = blockIdx.x * 256 + threadIdx.x;  // 0 .. 4*512*512-1
    const int b   = gid >> 18;
    const int pix = gid & (512 * 512 - 1);
    const int y   = pix >> 9, x = pix & 511;

    const unsigned idx = 0xFFFFFFFFu - (unsigned)(keys[b] & 0xFFFFFFFFull);
    const int iy = (int)(idx >> 7);      // idx / 128
    const int ix = (int)(idx & 127u);    // idx % 128

    // ---- bilinear upsample of first_masks (half-pixel centers, edge clamp) ----
    float mval[NCLS];
    {
        const float fsy = (float)y * 0.25f - 0.375f;
        const float fsx = (float)x * 0.25f - 0.375f;
        const int sy0 = (int)floorf(fsy), sx0 = (int)floorf(fsx);
        const float wy = fsy - (float)sy0, wx = fsx - (float)sx0;
        const int y0c = min(max(sy0, 0), 127), y1c = min(max(sy0 + 1, 0), 127);
        const int x0c = min(max(sx0, 0), 127), x1c = min(max(sx0 + 1, 0), 127);
        const float* fmb = first_masks + (size_t)b * NCLS * 16384u;
        #pragma unroll
        for (int k = 0; k < NCLS; ++k) {
            const float* p = fmb + (size_t)k * 16384u;
            const float v00 = p[y0c * 128 + x0c], v01 = p[y0c * 128 + x1c];
            const float v10 = p[y1c * 128 + x0c], v11 = p[y1c * 128 + x1c];
            mval[k] = (1.f - wy) * ((1.f - wx) * v00 + wx * v01)
                    +        wy  * ((1.f - wx) * v10 + wx * v11);
        }
    }

    // ---- circle region: compute hd_masks on the fly at gathered hd coords ----
    const int dyi = y - 256, dxi = x - 256;
    const float dyf = (float)dyi - 0.001f, dxf = (float)dxi - 0.001f;
    const float rad = sqrtf(dyf * dyf + dxf * dxf);
    if (rad <= 100.0f) {
        const int sy = iy * 4 + dyi, sx = ix * 4 + dxi;
        if (sy >= 0 && sy < 512 && sx >= 0 && sx < 512) {
            const int dy2 = sy - 256, dx2 = sx - 256;
            const float dyf2 = (float)dy2 - 0.001f, dxf2 = (float)dx2 - 0.001f;
            const float rad2 = sqrtf(dyf2 * dyf2 + dxf2 * dxf2);
            const float RMAXc = 1.41421356237309504880f * 256.0f - 250.0f;
            const float RTOT  = 250.0f + RMAXc;       // = sqrt(2)*256
            const float* il = im_large + (size_t)b * 3u * 2048u * 2048u;
            const float fy16 = (float)(iy * 16), fx16 = (float)(ix * 16);
            float hr = 0.f, hg = 0.f, hb = 0.f;
            if (rad2 > RTOT) {
                // outside everything -> zeros
            } else if (rad2 >= 250.0f) {
                ring_gather(il, 4, RTOT,   fy16, fx16, dyf2, dxf2, rad2, hr, hg, hb);
            } else if (rad2 >= 180.0f) {
                ring_gather(il, 3, 250.0f, fy16, fx16, dyf2, dxf2, rad2, hr, hg, hb);
            } else if (rad2 >= 100.0f) {
                ring_gather(il, 2, 180.0f, fy16, fx16, dyf2, dxf2, rad2, hr, hg, hb);
            } else {
                const int gy = iy * 16 + dy2, gx = ix * 16 + dx2;
                if (gy >= 0 && gy < 2048 && gx >= 0 && gx < 2048) {
                    hr = il[(size_t)0 * 2048u * 2048u + (size_t)gy * 2048u + gx];
                    hg = il[(size_t)1 * 2048u * 2048u + (size_t)gy * 2048u + gx];
                    hb = il[(size_t)2 * 2048u * 2048u + (size_t)gy * 2048u + gx];
                }
            }
            #pragma unroll
            for (int k = 0; k < NCLS; ++k)
                mval[k] = W2[k * 3 + 0] * hr + W2[k * 3 + 1] * hg
                        + W2[k * 3 + 2] * hb + b2[k];
        }
    }

    // ---- softmax over classes, write NCHW ----
    float mx = mval[0];
    #pragma unroll
    for (int k = 1; k < NCLS; ++k) mx = fmaxf(mx, mval[k]);
    float e[NCLS], sum = 0.f;
    #pragma unroll
    for (int k = 0; k < NCLS; ++k) { e[k] = expf(mval[k] - mx); sum += e[k]; }
    const float inv = 1.0f / sum;
    const size_t ob = (size_t)b * NCLS * 262144u + (size_t)y * 512u + x;
    #pragma unroll
    for (int k = 0; k < NCLS; ++k)
        out[ob + (size_t)k * 262144u] = e[k] * inv;
}

extern "C" void kernel_launch(void* const* d_in, const int* in_sizes, int n_in,
                              void* d_out, int out_size, void* d_ws, size_t ws_size,
                              hipStream_t stream) {
    const float* im       = (const float*)d_in[0];   // (4,3,512,512)
    const float* im_large = (const float*)d_in[1];   // (4,3,2048,2048)
    const float* W1       = (const float*)d_in[2];   // (11,3)
    const float* b1       = (const float*)d_in[3];   // (11,)
    const float* W2       = (const float*)d_in[4];   // (11,3)
    const float* b2       = (const float*)d_in[5];   // (11,)
    float* out            = (float*)d_out;           // (4,11,512,512)

    char* ws = (char*)d_ws;
    float* first_masks = (float*)ws;                                   // 4*11*128*128 f32
    unsigned long long* keys =
        (unsigned long long*)(ws + (size_t)4 * NCLS * 16384u * 4u);    // 4 x u64

    init_keys_kernel<<<1, 32, 0, stream>>>(keys);
    stage1_kernel<<<65536 / 256, 256, 0, stream>>>(im, W1, b1, first_masks, keys);
    stage3_kernel<<<(4 * 512 * 512) / 256, 256, 0, stream>>>(
        im_large, first_masks, keys, W2, b2, out);
}